// GCNConvCustom_58067957842338
// MI455X (gfx1250) — compile-verified
//
#include <hip/hip_runtime.h>
#include <hip/hip_bf16.h>

typedef float v2f __attribute__((ext_vector_type(2)));
typedef float v8f __attribute__((ext_vector_type(8)));

#define DDIM 256  // D_IN == D_OUT

// ---------------- degree kernels ----------------
__global__ void k_deg_init(unsigned* __restrict__ deg, int n) {
  int i = blockIdx.x * blockDim.x + threadIdx.x;
  if (i < n) deg[i] = 1u;  // self loop contributes 1 to every column sum
}

__global__ void k_deg_edges(const int* __restrict__ dst, unsigned* __restrict__ deg, int ne) {
  int e = blockIdx.x * blockDim.x + threadIdx.x;
  if (e < ne) atomicAdd(&deg[dst[e]], 1u);
}

__global__ void k_dinv(const unsigned* __restrict__ deg, float* __restrict__ dinv, int n) {
  int i = blockIdx.x * blockDim.x + threadIdx.x;
  if (i < n) dinv[i] = 1.0f / sqrtf((float)deg[i]);
}

// ---------------- Xs = diag(dinv) * (H @ W) via f32 WMMA ----------------
// Block: 512 threads = 16 waves. Block b owns rows [16b, 16b+16); wave w owns
// columns [16w, 16w+16). H stripe staged in LDS (16x256 f32 = 16KB).
__global__ __launch_bounds__(512)
void k_gemm_scale(const float* __restrict__ Hm, const float* __restrict__ Wm,
                  const float* __restrict__ dinv,
                  float* __restrict__ Xs, float* __restrict__ ACC) {
  __shared__ float lds[16 * DDIM];

  const int m0 = blockIdx.x * 16;

  // Coalesced cooperative load of 16x256 f32 tile (1024 float4, 2 per thread).
  const float4* src = (const float4*)(Hm + (size_t)m0 * DDIM);
  float4* dstl = (float4*)lds;
  dstl[threadIdx.x] = src[threadIdx.x];
  dstl[threadIdx.x + 512] = src[threadIdx.x + 512];
  __syncthreads();

  const int lane = threadIdx.x & 31;
  const int wave = threadIdx.x >> 5;
  const int hf = lane >> 4;   // 0: K=0,1   1: K=2,3
  const int r  = lane & 15;   // A: local row M;  B: local col N
  const int n0 = wave * 16;

  v8f c = {0.f, 0.f, 0.f, 0.f, 0.f, 0.f, 0.f, 0.f};

  #pragma unroll 4
  for (int k0 = 0; k0 < DDIM; k0 += 4) {
    // A frag (16x4): lane holds H[m0+r][k0+2*hf .. +1]  -> ds_load_b64
    v2f a = *(const v2f*)&lds[r * DDIM + k0 + 2 * hf];
    // B frag (4x16): lane holds W[k0+2*hf][n0+r], W[k0+2*hf+1][n0+r]
    v2f b;
    b.x = Wm[(k0 + 2 * hf) * DDIM + n0 + r];
    b.y = Wm[(k0 + 2 * hf + 1) * DDIM + n0 + r];
    c = __builtin_amdgcn_wmma_f32_16x16x4_f32(false, a, false, b, (short)0, c,
                                              false, false);
  }

  // C layout: VGPR v, lane l -> row = v + 8*(l>=16), col = l&15.
  // Write Xs = dinv[row] * X and seed ACC with the same (self-loop term).
  #pragma unroll
  for (int v = 0; v < 8; ++v) {
    int row = m0 + v + 8 * hf;
    float s = c[v] * dinv[row];
    size_t idx = (size_t)row * DDIM + n0 + r;
    Xs[idx] = s;
    ACC[idx] = s;
  }
}

// ---------------- edge scatter: ACC[src] += Xs[dst] ----------------
// One wave per edge: 2 coalesced b128 row loads + 8 fp32 atomic adds per lane.
__global__ __launch_bounds__(256)
void k_scatter(const int* __restrict__ el, int ne,
               const float* __restrict__ Xs, float* __restrict__ ACC) {
  int e = (blockIdx.x * blockDim.x + threadIdx.x) >> 5;
  int lane = threadIdx.x & 31;
  if (e >= ne) return;
  int s = el[e];        // row of A (output row)
  int d = el[ne + e];   // col of A (gather row)
  const float4* xrow = (const float4*)(Xs + (size_t)d * DDIM);
  float* arow = ACC + (size_t)s * DDIM;
  #pragma unroll
  for (int part = 0; part < 2; ++part) {
    float4 v = xrow[part * 32 + lane];
    int base = (part * 32 + lane) * 4;
    __hip_atomic_fetch_add(&arow[base + 0], v.x, __ATOMIC_RELAXED, __HIP_MEMORY_SCOPE_AGENT);
    __hip_atomic_fetch_add(&arow[base + 1], v.y, __ATOMIC_RELAXED, __HIP_MEMORY_SCOPE_AGENT);
    __hip_atomic_fetch_add(&arow[base + 2], v.z, __ATOMIC_RELAXED, __HIP_MEMORY_SCOPE_AGENT);
    __hip_atomic_fetch_add(&arow[base + 3], v.w, __ATOMIC_RELAXED, __HIP_MEMORY_SCOPE_AGENT);
  }
}

// ---------------- out = dinv[row] * ACC + bias[col] ----------------
__global__ void k_final(const float* __restrict__ ACC, const float* __restrict__ dinv,
                        const float* __restrict__ bias, float* __restrict__ out, int total) {
  int g = blockIdx.x * blockDim.x + threadIdx.x;
  if (g < total) out[g] = dinv[g >> 8] * ACC[g] + bias[g & (DDIM - 1)];
}

extern "C" void kernel_launch(void* const* d_in, const int* in_sizes, int n_in,
                              void* d_out, int out_size, void* d_ws, size_t ws_size,
                              hipStream_t stream) {
  const float* Hm   = (const float*)d_in[0];   // (N, 256) f32
  const float* Wm   = (const float*)d_in[1];   // (256, 256) f32
  const float* bias = (const float*)d_in[2];   // (256, 1) f32 -> flat 256
  const int*   el   = (const int*)d_in[3];     // (2, E) indices

  const int n  = in_sizes[0] / DDIM;  // 8192 nodes
  const int ne = in_sizes[3] / 2;     // 262144 edges
  float* out = (float*)d_out;

  // workspace layout: [deg u32 n][dinv f32 n][pad][Xs f32 n*256][ACC f32 n*256]
  char* ws = (char*)d_ws;
  unsigned* deg  = (unsigned*)ws;
  float*    dinv = (float*)(ws + (size_t)n * sizeof(unsigned));
  size_t off = (((size_t)n * 8) + 255) & ~(size_t)255;
  float* Xs  = (float*)(ws + off);
  float* ACC = Xs + (size_t)n * DDIM;

  k_deg_init<<<(n + 255) / 256, 256, 0, stream>>>(deg, n);
  k_deg_edges<<<(ne + 255) / 256, 256, 0, stream>>>(el + ne, deg, ne);
  k_dinv<<<(n + 255) / 256, 256, 0, stream>>>(deg, dinv, n);
  k_gemm_scale<<<n / 16, 512, 0, stream>>>(Hm, Wm, dinv, Xs, ACC);
  k_scatter<<<(ne * 32 + 255) / 256, 256, 0, stream>>>(el, ne, Xs, ACC);
  k_final<<<(n * DDIM + 255) / 256, 256, 0, stream>>>(ACC, dinv, bias, out, n * DDIM);
}